// RuleTransform_50680614093539
// MI455X (gfx1250) — compile-verified
//
#include <hip/hip_runtime.h>

typedef __attribute__((ext_vector_type(2))) float v2f;
typedef __attribute__((ext_vector_type(8))) float v8f;

static constexpr int kD = 256;          // hidden dim (matches reference)
static constexpr float kEps = 1e-12f;

// ---------------- setup: group tokens by rule (counting sort) ----------------

__global__ void rt_zero(int* counts, int* cursor, int nrules) {
  int t = blockIdx.x * blockDim.x + threadIdx.x;
  if (t < nrules) { counts[t] = 0; cursor[t] = 0; }
}

__global__ void rt_count(const int* __restrict__ token_ids,
                         const int* __restrict__ token_rules,
                         int* __restrict__ counts, int ntok) {
  int t = blockIdx.x * blockDim.x + threadIdx.x;
  if (t < ntok) atomicAdd(&counts[token_rules[token_ids[t]]], 1);
}

// Exclusive prefix scan over counts -> offsets[0..nrules].
// One block of 1024 threads; Hillis-Steele in LDS (nrules <= 1024 fast path).
__global__ __launch_bounds__(1024) void rt_scan(const int* __restrict__ counts,
                                                int* __restrict__ offsets,
                                                int nrules) {
  __shared__ int sh[1024];
  const int t = threadIdx.x;
  if (nrules <= 1024) {
    int v = (t < nrules) ? counts[t] : 0;
    sh[t] = v;
    __syncthreads();
#pragma unroll
    for (int ofs = 1; ofs < 1024; ofs <<= 1) {
      int x = (t >= ofs) ? sh[t - ofs] : 0;
      __syncthreads();
      sh[t] += x;
      __syncthreads();
    }
    if (t < nrules) offsets[t] = sh[t] - v;     // exclusive
    if (t == nrules - 1) offsets[nrules] = sh[t];
  } else if (t == 0) {                           // generic fallback
    int acc = 0;
    for (int i = 0; i < nrules; ++i) { offsets[i] = acc; acc += counts[i]; }
    offsets[nrules] = acc;
  }
}

__global__ void rt_scatter(const int* __restrict__ token_ids,
                           const int* __restrict__ token_rules,
                           const int* __restrict__ offsets,
                           int* __restrict__ cursor,
                           int* __restrict__ tlist, int ntok) {
  int t = blockIdx.x * blockDim.x + threadIdx.x;
  if (t < ntok) {
    int r = token_rules[token_ids[t]];
    int pos = offsets[r] + atomicAdd(&cursor[r], 1);
    tlist[pos] = t;
  }
}

// ---------------- main: per-rule GEMM with fused Frobenius norm ----------------
// Block = 256 threads (8 wave32). Wave w owns output columns [32w, 32w+32).
// A (<=16 tokens x 256) staged in LDS, shared by all waves.
// B fragments streamed non-temporally from the rule matrix (read exactly once).

__global__ __launch_bounds__(256) void rt_gemm(
    const float* __restrict__ hidden,
    const float* __restrict__ rules,
    const int* __restrict__ offsets,
    const int* __restrict__ tlist,
    float* __restrict__ out) {
  const int r = blockIdx.x;
  const int start = offsets[r];
  const int cnt = offsets[r + 1] - start;
  if (cnt == 0) return;

  const int tid = threadIdx.x;
  const int lane = tid & 31;
  const int wave = tid >> 5;
  const int h = lane >> 4;          // lane half (0: lanes 0-15, 1: lanes 16-31)
  const int col = lane & 15;
  const int colbase = wave * 32;    // two 16-wide tiles: colbase, colbase+16

  __shared__ float Ash[16 * 260];   // stride 260 (=4 mod 64 banks) avoids conflicts
  __shared__ int   toksh[16];
  __shared__ float redsh[8];
  __shared__ float scalesh;

  const float* __restrict__ M = rules + (size_t)r * (kD * kD);
  float scale = 0.0f;

  for (int c0 = 0; c0 < cnt; c0 += 16) {
    __syncthreads();  // previous chunk fully done with LDS
    if (tid < 16) toksh[tid] = (c0 + tid < cnt) ? tlist[start + c0 + tid] : -1;
    __syncthreads();

    // cooperative A-chunk load: 16 rows x 64 float4 (pad rows with zeros)
    for (int i = tid; i < 16 * 64; i += 256) {
      int m = i >> 6;
      int seg = i & 63;
      int tok = toksh[m];
      float4 v = make_float4(0.f, 0.f, 0.f, 0.f);
      if (tok >= 0)
        v = reinterpret_cast<const float4*>(hidden + (size_t)tok * kD)[seg];
      *reinterpret_cast<float4*>(&Ash[m * 260 + seg * 4]) = v;
    }
    __syncthreads();

    v8f acc0 = {};
    v8f acc1 = {};
    float sumsq = 0.0f;

    const float* Mc = M + colbase + col;
#pragma unroll 4
    for (int k0 = 0; k0 < kD; k0 += 4) {
      // A fragment: lane=m(=col), vgpr v holds K = k0 + 2h + v
      v2f a = *reinterpret_cast<const v2f*>(&Ash[col * 260 + k0 + 2 * h]);
      // B fragment (K = vgpr + 2*half, mirroring C/D row striping)
      const float* p = Mc + (size_t)(k0 + 2 * h) * kD;
      v2f b0, b1;
      b0.x = __builtin_nontemporal_load(p);
      b0.y = __builtin_nontemporal_load(p + kD);
      b1.x = __builtin_nontemporal_load(p + 16);
      b1.y = __builtin_nontemporal_load(p + kD + 16);
      sumsq += b0.x * b0.x + b0.y * b0.y + b1.x * b1.x + b1.y * b1.y;
      acc0 = __builtin_amdgcn_wmma_f32_16x16x4_f32(false, a, false, b0,
                                                   (short)0, acc0, false, false);
      acc1 = __builtin_amdgcn_wmma_f32_16x16x4_f32(false, a, false, b1,
                                                   (short)0, acc1, false, false);
    }

    // Frobenius norm: each wave saw its own 32 columns exactly once -> block sum
    if (c0 == 0) {
      float s = sumsq;
      s += __shfl_xor(s, 16);
      s += __shfl_xor(s, 8);
      s += __shfl_xor(s, 4);
      s += __shfl_xor(s, 2);
      s += __shfl_xor(s, 1);
      if (lane == 0) redsh[wave] = s;
      __syncthreads();
      if (tid == 0) {
        float t = 0.f;
#pragma unroll
        for (int i = 0; i < 8; ++i) t += redsh[i];
        scalesh = 1.0f / fmaxf(sqrtf(t), kEps);
      }
      __syncthreads();
      scale = scalesh;
    }

#pragma unroll
    for (int i = 0; i < 8; ++i) { acc0[i] *= scale; acc1[i] *= scale; }

    // C/D layout: vgpr rr, row = rr + 8*half, col = lane&15
#pragma unroll
    for (int rr = 0; rr < 8; ++rr) {
      int row = rr + 8 * h;
      int tok = toksh[row];
      if (tok >= 0) {
        float* o = out + (size_t)tok * kD + colbase + col;
        __builtin_nontemporal_store(acc0[rr], o);
        __builtin_nontemporal_store(acc1[rr], o + 16);
      }
    }
  }
}

// ---------------- launch ----------------

extern "C" void kernel_launch(void* const* d_in, const int* in_sizes, int n_in,
                              void* d_out, int out_size, void* d_ws, size_t ws_size,
                              hipStream_t stream) {
  const float* hidden      = (const float*)d_in[0];
  const int*   token_ids   = (const int*)d_in[1];
  const float* rules       = (const float*)d_in[2];
  const int*   token_rules = (const int*)d_in[3];
  float*       out         = (float*)d_out;

  const int ntok   = in_sizes[1];               // B*S = 2048
  const int nrules = in_sizes[2] / (kD * kD);   // 1024

  int* counts  = (int*)d_ws;            // nrules
  int* offsets = counts + nrules;       // nrules + 1
  int* cursor  = offsets + nrules + 1;  // nrules
  int* tlist   = cursor + nrules;       // ntok

  rt_zero<<<(nrules + 255) / 256, 256, 0, stream>>>(counts, cursor, nrules);
  rt_count<<<(ntok + 255) / 256, 256, 0, stream>>>(token_ids, token_rules, counts, ntok);
  rt_scan<<<1, 1024, 0, stream>>>(counts, offsets, nrules);
  rt_scatter<<<(ntok + 255) / 256, 256, 0, stream>>>(token_ids, token_rules,
                                                     offsets, cursor, tlist, ntok);
  rt_gemm<<<nrules, 256, 0, stream>>>(hidden, rules, offsets, tlist, out);
}